// WindowAttentionWithLoRA_72593537237691
// MI455X (gfx1250) — compile-verified
//
#include <hip/hip_runtime.h>

typedef __attribute__((ext_vector_type(2))) float v2f;
typedef __attribute__((ext_vector_type(8))) float v8f;

#define DIM        512
#define THREE_DIM  1536
#define NUM_HEADS  16
#define RANK       4
#define HEAD_DIM   32
#define NTOK       49
#define NWIN       2048
#define NMASK      64
#define MROWS      (NWIN * NTOK)   /* 100352, divisible by 64 */
#define ATTN_SCALE 0.17677669529663687f   /* 32^-0.5 */

static __device__ __forceinline__ int imin(int a, int b) { return a < b ? a : b; }

// ---------------------------------------------------------------------------
// Kernel 0: fold LoRA into effective QKV weight: Weff = Wqkv + Bup @ Adown
// ---------------------------------------------------------------------------
__global__ void fold_lora(const float* __restrict__ Wqkv,
                          const float* __restrict__ Adown,
                          const float* __restrict__ Bup,
                          float* __restrict__ Weff) {
    int idx = blockIdx.x * blockDim.x + threadIdx.x;
    if (idx >= THREE_DIM * DIM) return;
    int o = idx / DIM;
    int i = idx - o * DIM;
    float acc = Wqkv[idx];
#pragma unroll
    for (int r = 0; r < RANK; ++r)
        acc += Bup[o * RANK + r] * Adown[r * DIM + i];
    Weff[idx] = acc;
}

// ---------------------------------------------------------------------------
// NT GEMM with bias: D[M x N] = A[M x K] * W[N x K]^T + bias[N]
// One wave per 64x64 output block: 4x4 register-blocked 16x16 WMMA tiles.
// Per K-step (K += 4): 8 float2 loads feed 16 v_wmma_f32_16x16x4_f32
// (0.5 VMEM ops per WMMA, 4x less L2 traffic than unblocked).
// A-fragment layout (ISA 7.12.2, 32-bit A 16x4): row M = lane&15,
// lanes 0-15 hold K=k0,k0+1; lanes 16-31 hold K=k0+2,k0+3.
// ---------------------------------------------------------------------------
__global__ __launch_bounds__(256) void gemm_nt_bias(const float* __restrict__ A,
                                                    const float* __restrict__ W,
                                                    const float* __restrict__ bias,
                                                    float* __restrict__ D,
                                                    int MB, int NB, int K, int ldD) {
    int lane = threadIdx.x & 31;
    int wv   = threadIdx.x >> 5;
    int gw   = blockIdx.x * (blockDim.x >> 5) + wv;
    int bM = gw / NB;
    int bN = gw - bM * NB;
    if (bM >= MB) return;                 // wave-uniform guard (EXEC stays all-1)

    int half = lane >> 4;
    int l    = lane & 15;

    const float* ap[4];
    const float* wp[4];
#pragma unroll
    for (int i = 0; i < 4; ++i)
        ap[i] = A + (size_t)(bM * 64 + i * 16 + l) * K + 2 * half;
#pragma unroll
    for (int j = 0; j < 4; ++j)
        wp[j] = W + (size_t)(bN * 64 + j * 16 + l) * K + 2 * half;

    v8f acc[16] = {};
    for (int k0 = 0; k0 < K; k0 += 4) {
        v2f a[4], b[4];
#pragma unroll
        for (int i = 0; i < 4; ++i) a[i] = *(const v2f*)(ap[i] + k0);
#pragma unroll
        for (int j = 0; j < 4; ++j) b[j] = *(const v2f*)(wp[j] + k0);
#pragma unroll
        for (int i = 0; i < 4; ++i)
#pragma unroll
            for (int j = 0; j < 4; ++j)
                acc[i * 4 + j] = __builtin_amdgcn_wmma_f32_16x16x4_f32(
                    false, a[i], false, b[j], (short)0, acc[i * 4 + j], false, false);
    }

#pragma unroll
    for (int j = 0; j < 4; ++j) {
        int col  = bN * 64 + j * 16 + l;
        float bc = bias[col];
#pragma unroll
        for (int i = 0; i < 4; ++i) {
#pragma unroll
            for (int r = 0; r < 8; ++r) {
                int row = bM * 64 + i * 16 + r + 8 * half;
                D[(size_t)row * ldD + col] = acc[i * 4 + j][r] + bc;
            }
        }
    }
}

// ---------------------------------------------------------------------------
// Kernel 2: window attention, one wave per (window b, head h).
//   S = (Q K^T) * scale + mask   (64x64 padded in LDS, invalid -> -1e30)
//   P = softmax_row(S)           (cols >= 49 zeroed)
//   O = P V                      (K padded to 64; P[:,49:]==0 kills garbage V)
// Q-fragments hoisted across tj (4x reuse); V-fragments hoisted across ti.
// ---------------------------------------------------------------------------
__global__ __launch_bounds__(128) void window_attn(const float* __restrict__ qkv,
                                                   const float* __restrict__ mask,
                                                   float* __restrict__ AO) {
    __shared__ float smem[4 * 64 * 64];   // 64 KB: one 64x64 f32 tile per wave
    int lane = threadIdx.x & 31;
    int wv   = threadIdx.x >> 5;
    float* S = smem + wv * 64 * 64;

    int gid = blockIdx.x * 4 + wv;
    int b = gid >> 4;
    int h = gid & 15;
    int half = lane >> 4;
    int l    = lane & 15;

    const size_t base = (size_t)b * NTOK * THREE_DIM + (size_t)h * HEAD_DIM;
    const float* Q  = qkv + base;
    const float* Kp = qkv + base + DIM;
    const float* V  = qkv + base + 2 * DIM;
    const float* mwin = mask + (size_t)(b & (NMASK - 1)) * NTOK * NTOK;

    // ---- S = Q K^T (4x4 tiles of 16x16, K = HEAD_DIM = 32) ----
    for (int ti = 0; ti < 4; ++ti) {
        int rq = imin(ti * 16 + l, NTOK - 1);             // clamped: stays in-bounds
        const float* qp = Q + (size_t)rq * THREE_DIM + 2 * half;
        v2f qf[8];
#pragma unroll
        for (int k = 0; k < 8; ++k) qf[k] = *(const v2f*)(qp + 4 * k);

        for (int tj = 0; tj < 4; ++tj) {
            int rk = imin(tj * 16 + l, NTOK - 1);
            const float* kp = Kp + (size_t)rk * THREE_DIM + 2 * half;
            v8f acc = {};
#pragma unroll
            for (int k = 0; k < 8; ++k) {
                v2f bv = *(const v2f*)(kp + 4 * k);
                acc = __builtin_amdgcn_wmma_f32_16x16x4_f32(false, qf[k], false, bv,
                                                            (short)0, acc, false, false);
            }
            int jj = tj * 16 + l;
#pragma unroll
            for (int j = 0; j < 8; ++j) {
                int ii = ti * 16 + j + 8 * half;
                float val = -1e30f;
                if (ii < NTOK && jj < NTOK)
                    val = acc[j] * ATTN_SCALE + mwin[ii * NTOK + jj];
                S[ii * 64 + jj] = val;
            }
        }
    }
    __syncthreads();

    // ---- row softmax: each lane owns 2 rows ----
#pragma unroll
    for (int rr = 0; rr < 2; ++rr) {
        int r = lane + rr * 32;
        float* srow = S + r * 64;
        if (r < NTOK) {
            float m = -1e30f;
            for (int j = 0; j < NTOK; ++j) m = fmaxf(m, srow[j]);
            float s = 0.f;
            for (int j = 0; j < NTOK; ++j) { float e = __expf(srow[j] - m); srow[j] = e; s += e; }
            float inv = 1.f / s;
            for (int j = 0; j < NTOK; ++j) srow[j] *= inv;
            for (int j = NTOK; j < 64; ++j) srow[j] = 0.f;
        } else {
            for (int j = 0; j < 64; ++j) srow[j] = 0.f;
        }
    }
    __syncthreads();

    // ---- O = P V : K = 64 (padded; P cols >=49 are zero) ----
    for (int tj = 0; tj < 2; ++tj) {
        int d = tj * 16 + l;
        v2f vf[16];
#pragma unroll
        for (int k = 0; k < 16; ++k) {
            int r0 = imin(4 * k + 2 * half,     NTOK - 1);
            int r1 = imin(4 * k + 2 * half + 1, NTOK - 1);
            vf[k].x = V[(size_t)r0 * THREE_DIM + d];
            vf[k].y = V[(size_t)r1 * THREE_DIM + d];
        }
        for (int ti = 0; ti < 4; ++ti) {
            int ip = ti * 16 + l;
            v8f acc = {};
#pragma unroll
            for (int k = 0; k < 16; ++k) {
                v2f av = *(const v2f*)(S + ip * 64 + 4 * k + 2 * half);
                acc = __builtin_amdgcn_wmma_f32_16x16x4_f32(false, av, false, vf[k],
                                                            (short)0, acc, false, false);
            }
#pragma unroll
            for (int j = 0; j < 8; ++j) {
                int m = ti * 16 + j + 8 * half;
                if (m < NTOK)
                    AO[((size_t)b * NTOK + m) * DIM + h * HEAD_DIM + d] = acc[j];
            }
        }
    }
}

// ---------------------------------------------------------------------------
extern "C" void kernel_launch(void* const* d_in, const int* in_sizes, int n_in,
                              void* d_out, int out_size, void* d_ws, size_t ws_size,
                              hipStream_t stream) {
    const float* x     = (const float*)d_in[0];
    const float* mask  = (const float*)d_in[1];
    const float* Wqkv  = (const float*)d_in[2];
    const float* bqkv  = (const float*)d_in[3];
    const float* Adown = (const float*)d_in[4];
    const float* Bup   = (const float*)d_in[5];
    const float* Wproj = (const float*)d_in[6];
    const float* bproj = (const float*)d_in[7];
    float* out = (float*)d_out;

    char* ws = (char*)d_ws;
    const size_t weff_bytes = (size_t)THREE_DIM * DIM * sizeof(float);          // 3 MB
    const size_t qkv_bytes  = (size_t)MROWS * THREE_DIM * sizeof(float);        // ~616 MB
    float* Weff = (float*)ws;
    float* qkv  = (float*)(ws + weff_bytes);
    float* AO   = (float*)(ws + weff_bytes + qkv_bytes);                        // ~205 MB

    // 0) fold LoRA into the QKV weight
    fold_lora<<<(THREE_DIM * DIM + 255) / 256, 256, 0, stream>>>(Wqkv, Adown, Bup, Weff);

    // 1) QKV GEMM: [100352 x 512] @ [1536 x 512]^T + b  -> qkv
    {
        int MB = MROWS / 64, NB = THREE_DIM / 64;         // 1568 x 24
        int waves = MB * NB;                              // 37632
        gemm_nt_bias<<<(waves + 7) / 8, 256, 0, stream>>>(x, Weff, bqkv, qkv,
                                                          MB, NB, DIM, THREE_DIM);
    }

    // 2) per-(window, head) attention
    window_attn<<<(NWIN * NUM_HEADS) / 4, 128, 0, stream>>>(qkv, mask, AO);

    // 3) output projection: [100352 x 512] @ [512 x 512]^T + b -> out
    {
        int MB = MROWS / 64, NB = DIM / 64;               // 1568 x 8
        int waves = MB * NB;                              // 12544
        gemm_nt_bias<<<(waves + 7) / 8, 256, 0, stream>>>(AO, Wproj, bproj, out,
                                                          MB, NB, DIM, DIM);
    }
}